// GetAttenBias_63299228009184
// MI455X (gfx1250) — compile-verified
//
#include <hip/hip_runtime.h>
#include <stdint.h>

// GetAttenBias for MI455X (gfx1250): degree-embedding fuse.
// Bandwidth-bound (~250MB @ 23.3 TB/s ≈ 11us floor). No matmul exists in the
// op, so WMMA is inapplicable; the MI455X levers are: b128 vector VMEM,
// NT cache hints on the read-once/write-once streams (keep the hot embedding
// rows resident), wave-uniform gathers (wave32: one node per wave), 64-bit
// atomic hash-set dedup, and global_prefetch_b8 on the streamed edge list.

#define N_NODESC 100000
#define N_EDGESC 1600000
#define DIMC     256
#define V4C      (DIMC / 4)   // 64 float4 per node
#define MAXBIN   511

typedef unsigned long long u64;
typedef float v4f __attribute__((ext_vector_type(4)));  // raw vector: OK for
                                                        // __builtin_nontemporal_*
#define EMPTY_KEY 0xFFFFFFFFFFFFFFFFull

// ---------------------------------------------------------------------------
// Kernel 1: re-initialize scratch every call (degrees = 0, hash slots = EMPTY)
// ---------------------------------------------------------------------------
__global__ void ws_init_kernel(int* __restrict__ deg, u64* __restrict__ table,
                               int table_size) {
  int gid    = blockIdx.x * blockDim.x + threadIdx.x;
  int stride = gridDim.x * blockDim.x;
  for (int i = gid; i < 2 * N_NODESC; i += stride) deg[i] = 0;
  for (int i = gid; i < table_size; i += stride) table[i] = EMPTY_KEY;
}

// ---------------------------------------------------------------------------
// Kernel 2: dedup edges via atomicCAS hash set; count unique-pair degrees.
// Key = src*N + dst < 2^37, so EMPTY_KEY is unreachable as a real key.
// Slots only transition EMPTY -> key and never change after, so each unique
// key commits to exactly one slot regardless of thread timing ->
// degree counts are deterministic run-to-run.
// ---------------------------------------------------------------------------
__global__ void dedup_degree_kernel(const int* __restrict__ ei,
                                    int* __restrict__ in_deg,
                                    int* __restrict__ out_deg,
                                    u64* __restrict__ table,
                                    unsigned table_mask) {
  int e = blockIdx.x * blockDim.x + threadIdx.x;
  if (e >= N_EDGESC) return;

  // gfx1250 explicit prefetch (global_prefetch_b8) on the streamed index rows.
  if (e + 8192 < N_EDGESC) {
    __builtin_prefetch(ei + e + 8192, 0, 1);
    __builtin_prefetch(ei + N_EDGESC + e + 8192, 0, 1);
  }

  int src = ei[e];
  int dst = ei[N_EDGESC + e];
  if ((unsigned)src >= (unsigned)N_NODESC || (unsigned)dst >= (unsigned)N_NODESC)
    return;  // defensive; reference indices are in-range

  u64 key  = (u64)src * (u64)N_NODESC + (u64)dst;
  u64 h    = key * 0x9E3779B97F4A7C15ull;
  unsigned slot = (unsigned)(h >> 32) & table_mask;

  for (;;) {
    u64 prev = atomicCAS(&table[slot], EMPTY_KEY, key);
    if (prev == EMPTY_KEY) {  // we inserted: first sighting of this pair
      atomicAdd(&in_deg[src], 1);
      atomicAdd(&out_deg[dst], 1);
      return;
    }
    if (prev == key) return;         // duplicate pair, count once only
    slot = (slot + 1) & table_mask;  // linear probe (slots are write-once)
  }
}

// ---------------------------------------------------------------------------
// Kernel 3: out = x + in_table[in_deg]*(in_deg>0) + out_table[out_deg]*(out_deg>0)
// One node per wave32 (node = gid>>5): degree loads are wave-uniform (single
// cacheline broadcast), and each lane does 2x b128 -> two coalesced 512B
// segments per wave. x loads and out stores are NT (read/written exactly
// once); table gathers use default RT policy so hot rows stay cached.
// ---------------------------------------------------------------------------
__global__ void fuse_kernel(const v4f* __restrict__ x4,
                            const v4f* __restrict__ tin4,
                            const v4f* __restrict__ tout4,
                            const int* __restrict__ in_deg,
                            const int* __restrict__ out_deg,
                            v4f* __restrict__ out4) {
  int gid = blockIdx.x * blockDim.x + threadIdx.x;
  const int total = N_NODESC * 32;  // 3.2M lanes, 2 float4 each
  if (gid >= total) return;

  int node = gid >> 5;        // one wave32 == one node
  int d4   = gid & 31;        // lane's first float4 within the node
  int base = node * V4C + d4; // second float4 at base+32

  int id = in_deg[node];
  int od = out_deg[node];
  if (id > MAXBIN) id = MAXBIN;   // JAX clamp semantics for OOB gather
  if (od > MAXBIN) od = MAXBIN;

  v4f v0 = __builtin_nontemporal_load(&x4[base]);
  v4f v1 = __builtin_nontemporal_load(&x4[base + 32]);

  if (id > 0) {                   // padding_idx=0 -> contributes zero; skip load
    v0 += tin4[id * V4C + d4];
    v1 += tin4[id * V4C + d4 + 32];
  }
  if (od > 0) {
    v0 += tout4[od * V4C + d4];
    v1 += tout4[od * V4C + d4 + 32];
  }
  __builtin_nontemporal_store(v0, &out4[base]);
  __builtin_nontemporal_store(v1, &out4[base + 32]);
}

// ---------------------------------------------------------------------------
// Kernel 4: zero the tuple tail (attn_bias scalar) of d_out, if present.
// ---------------------------------------------------------------------------
__global__ void zero_tail_kernel(float* __restrict__ out, int start, int total) {
  int i = start + blockIdx.x * blockDim.x + threadIdx.x;
  if (i < total) out[i] = 0.0f;
}

// ---------------------------------------------------------------------------
extern "C" void kernel_launch(void* const* d_in, const int* in_sizes, int n_in,
                              void* d_out, int out_size, void* d_ws, size_t ws_size,
                              hipStream_t stream) {
  (void)in_sizes; (void)n_in;
  const float* x    = (const float*)d_in[0];
  // d_in[1] = edge_feature: unused by the reference output
  const int*   ei   = (const int*)d_in[2];   // [2, E] flat, per harness int cast
  const float* tin  = (const float*)d_in[3];
  const float* tout = (const float*)d_in[4];
  float*       out  = (float*)d_out;

  // Workspace layout: [in_deg: N int][out_deg: N int][hash table: 2^logT u64]
  int*   deg       = (int*)d_ws;
  size_t deg_bytes = (size_t)(2 * N_NODESC) * sizeof(int);  // 800000B, 8B-aligned
  int logT = 22;                                            // prefer 4M slots (32MB)
  while (logT > 16 && deg_bytes + (8ull << logT) > ws_size) --logT;
  int      table_size = 1 << logT;
  unsigned table_mask = (unsigned)(table_size - 1);
  u64*     table      = (u64*)((char*)d_ws + deg_bytes);

  ws_init_kernel<<<4096, 256, 0, stream>>>(deg, table, table_size);

  dedup_degree_kernel<<<(N_EDGESC + 255) / 256, 256, 0, stream>>>(
      ei, deg, deg + N_NODESC, table, table_mask);

  const int lanes = N_NODESC * 32;  // 3,200,000 lanes (2 float4 each)
  fuse_kernel<<<(lanes + 255) / 256, 256, 0, stream>>>(
      (const v4f*)x, (const v4f*)tin, (const v4f*)tout,
      deg, deg + N_NODESC, (v4f*)out);

  const int nf_elems = N_NODESC * DIMC;  // 25,600,000
  if (out_size > nf_elems) {
    int tail = out_size - nf_elems;
    zero_tail_kernel<<<(tail + 255) / 256, 256, 0, stream>>>(out, nf_elems, out_size);
  }
}